// BMN_79001628442763
// MI455X (gfx1250) — compile-verified
//
#include <hip/hip_runtime.h>
#include <hip/hip_bf16.h>

// ---------------- problem constants ----------------
#define T_    100
#define DM_   100
#define NS_   32
#define FEAT_ 400
#define H1_   256
#define H2_   128
#define H3_   512
#define B_    2
#define NT_   10000   // DM_*T_
#define K2_   3200    // NS_*T_
#define TP_   112     // padded T (multiple of 16) for branch-free GEMM-G

typedef __attribute__((ext_vector_type(2))) float v2f;
typedef __attribute__((ext_vector_type(8))) float v8f;
typedef __attribute__((ext_vector_type(4))) int   v4i;

// gfx1250 async global->LDS copy path (guarded; falls back to plain LDS stores)
#if defined(__has_builtin)
# if __has_builtin(__builtin_amdgcn_global_load_async_to_lds_b128) && \
     __has_builtin(__builtin_amdgcn_s_wait_asynccnt)
#  define USE_ASYNC_LDS 1
# endif
#endif
#ifndef USE_ASYNC_LDS
# define USE_ASYNC_LDS 0
#endif
#define AS1_ __attribute__((address_space(1)))
#define AS3_ __attribute__((address_space(3)))

// f32 WMMA: D(16x16) += A(16x4) * B(4x16), wave32, one tile per wave.
__device__ __forceinline__ v8f wmma_f32_4(v2f a, v2f b, v8f c) {
  return __builtin_amdgcn_wmma_f32_16x16x4_f32(
      false, a, false, b, (short)0, c, false, false);
}

__device__ __forceinline__ float sigmoidf_(float v) {
  return 1.0f / (1.0f + __expf(-v));
}

// ---------------------------------------------------------------
__global__ __launch_bounds__(256) void k_zero(float* __restrict__ p, int n) {
  int i = blockIdx.x * 256 + threadIdx.x;
  if (i < n) p[i] = 0.f;
}

// ---------------------------------------------------------------
// Grouped conv1d (k=3, pad=1, groups=4, 256 out ch) + ReLU.
// grid = B_*T_, block = 256 (one thread per out channel).
// ---------------------------------------------------------------
__global__ __launch_bounds__(256) void k_conv1d_grp_relu(
    const float* __restrict__ in, const float* __restrict__ w,
    const float* __restrict__ bias, float* __restrict__ out, int cin_pg) {
  int bt = blockIdx.x;
  int t = bt % T_, b = bt / T_;
  int oc = threadIdx.x;
  int g = oc >> 6;
  const float* wp = w + (size_t)oc * cin_pg * 3;
  const float* ip = in + ((size_t)b * (4 * cin_pg) + (size_t)g * cin_pg) * T_ + t;
  float acc = bias[oc];
  for (int ic = 0; ic < cin_pg; ++ic) {
    const float* r = ip + (size_t)ic * T_;
    float w0 = wp[0], w1 = wp[1], w2 = wp[2];
    wp += 3;
    if (t > 0)       acc += w0 * r[-1];
    acc += w1 * r[0];
    if (t < T_ - 1)  acc += w2 * r[1];
  }
  out[((size_t)b * H1_ + oc) * T_ + t] = acc > 0.f ? acc : 0.f;
}

// ---------------------------------------------------------------
// Start / End head.
// ---------------------------------------------------------------
__global__ __launch_bounds__(256) void k_head(
    const float* __restrict__ h, const float* __restrict__ w1,
    const float* __restrict__ b1, const float* __restrict__ w2,
    const float* __restrict__ b2, float* __restrict__ outvec) {
  __shared__ float red[256];
  int bt = blockIdx.x;
  int t = bt % T_, b = bt / T_;
  int oc = threadIdx.x;
  int g = oc >> 6;
  const float* wp = w1 + (size_t)oc * 64 * 3;
  const float* ip = h + ((size_t)b * H1_ + (size_t)g * 64) * T_ + t;
  float acc = b1[oc];
  for (int ic = 0; ic < 64; ++ic) {
    const float* r = ip + (size_t)ic * T_;
    float w0 = wp[0], wc = wp[1], w2_ = wp[2];
    wp += 3;
    if (t > 0)      acc += w0 * r[-1];
    acc += wc * r[0];
    if (t < T_ - 1) acc += w2_ * r[1];
  }
  acc = acc > 0.f ? acc : 0.f;
  red[oc] = acc * w2[oc];
  __syncthreads();
  for (int st = 128; st > 0; st >>= 1) {
    if (oc < st) red[oc] += red[oc + st];
    __syncthreads();
  }
  if (oc == 0) outvec[b * T_ + t] = sigmoidf_(red[0] + b2[0]);
}

// ---------------------------------------------------------------
// Dense conv1d (256->256, k=3, pad=1) + ReLU -> zero-padded p [B][H1][TP_].
// ---------------------------------------------------------------
__global__ __launch_bounds__(256) void k_conv1d_dense_relu(
    const float* __restrict__ in, const float* __restrict__ w,
    const float* __restrict__ bias, float* __restrict__ out) {
  __shared__ float sx[3][H1_];
  int bt = blockIdx.x;
  int t = bt % T_, b = bt / T_;
  int tid = threadIdx.x;
  #pragma unroll
  for (int k = 0; k < 3; ++k) {
    int tt = t + k - 1;
    sx[k][tid] = (tt >= 0 && tt < T_) ? in[((size_t)b * H1_ + tid) * T_ + tt] : 0.f;
  }
  __syncthreads();
  const float* wp = w + (size_t)tid * H1_ * 3;
  float acc = bias[tid];
  for (int ic = 0; ic < H1_; ++ic) {
    acc += wp[0] * sx[0][ic] + wp[1] * sx[1][ic] + wp[2] * sx[2][ic];
    wp += 3;
  }
  out[((size_t)b * H1_ + tid) * TP_ + t] = acc > 0.f ? acc : 0.f;
}

// ---------------------------------------------------------------
// GEMM-G (WMMA): Gt[b][n*T_+t][o] = sum_c w3d[o][c][n] * p[b][c][t]
// K-major output so GEMM-FMAP's A fragments are lane-contiguous.
// B operand reads padded p (branch-free K loop).
// grid = (B_*NS_*32*7)/4 blocks of 128 threads (4 waves).
// ---------------------------------------------------------------
__global__ __launch_bounds__(128) void k_gemm_g(
    const float* __restrict__ w3d, const float* __restrict__ p,
    float* __restrict__ Gt) {
  const int MT = H3_ / 16;   // 32
  const int NTIL = 7;        // ceil(100/16)
  int wave = threadIdx.x >> 5, lane = threadIdx.x & 31;
  int tile = blockIdx.x * 4 + wave;
  int nt = tile % NTIL; tile /= NTIL;
  int mt = tile % MT;   tile /= MT;
  int n  = tile % NS_;  tile /= NS_;
  int b  = tile;
  int o0 = mt * 16, j0 = nt * 16;
  int mrow = lane & 15;
  int koff = (lane >> 4) << 1;
  int jg = j0 + (lane & 15);
  v8f acc = {};
  const float* A  = w3d + (size_t)o0 * (H1_ * NS_) + n;
  const float* Bm = p + (size_t)b * (H1_ * TP_);
  for (int k0 = 0; k0 < H1_; k0 += 4) {
    v2f a, bb;
    const float* Ap = A + (size_t)mrow * (H1_ * NS_) + (size_t)(k0 + koff) * NS_;
    a.x = Ap[0];
    a.y = Ap[NS_];
    const float* Bp = Bm + (size_t)(k0 + koff) * TP_ + jg;
    bb.x = Bp[0];
    bb.y = Bp[TP_];
    acc = wmma_f32_4(a, bb, acc);
  }
  if (jg < T_) {
    float* gp = Gt + ((size_t)b * K2_ + (size_t)n * T_ + jg) * H3_ + o0 + ((lane >> 4) << 3);
    float4 lo = {acc[0], acc[1], acc[2], acc[3]};
    float4 hi = {acc[4], acc[5], acc[6], acc[7]};
    *(float4*)gp       = lo;
    *(float4*)(gp + 4) = hi;
  }
}

// ---------------------------------------------------------------
// GEMM-FMAP (WMMA + async LDS staging): per batch
//   fmap[o][j] = relu( sum_kk Gt[b][kk][o] * mask[t(kk)][n(kk)][j] + b3d[o] )
// M=512, N=10000, K=3200.  Block = 256 thr (8 waves) covers M=512 x N=32:
// waves = 4 M-groups (128 rows, 8 tiles each) x 2 N-groups (16 cols).
// Mask tile (32K x 32N) double-buffered in LDS via async b128 copies ->
// mask is read exactly once per batch.
// grid = B_ * 313 blocks (last block is a guarded tail).
// ---------------------------------------------------------------
#define KC_   32
#define JB_   32
#define NJB_  313   // ceil(NT_/JB_)
__global__ __launch_bounds__(256) void k_gemm_fmap(
    const float* __restrict__ Gt, const float* __restrict__ mask,
    const float* __restrict__ b3d, float* __restrict__ fmap) {
  __shared__ float sB[2][KC_][JB_];
  int tid = threadIdx.x;
  int wave = tid >> 5, lane = tid & 31;
  int blk = blockIdx.x;
  int jblk = blk % NJB_;
  int b = blk / NJB_;
  int j0 = jblk * JB_;
  bool tail = (j0 + JB_ > NT_);

  int wgM = wave >> 1;                 // 0..3 -> M base = wgM*128
  int ngN = wave & 1;                  // 0..1 -> N base = j0 + ngN*16
  int mrow = lane & 15;
  int koff = (lane >> 4) << 1;
  int jcol = lane & 15;
  int jg = j0 + ngN * 16 + jcol;

  // staging lane map: 256 threads cover KC_ rows x (JB_/4) float4 columns
  int rB = tid >> 3;                   // 0..31 : row within chunk
  int cB = (tid & 7) << 2;             // 0,4,..,28 : float4 column

  v8f acc[8];
  #pragma unroll
  for (int i = 0; i < 8; ++i) acc[i] = (v8f){};

  auto stage = [&](int buf, int kbase) {
    int kk = kbase + rB;
    int n = kk / T_;
    int t = kk - n * T_;
    int jj = j0 + cB;
    const float* gp = mask + (size_t)t * (NS_ * NT_) + (size_t)n * NT_ + jj;
    if (!tail) {
#if USE_ASYNC_LDS
      __builtin_amdgcn_global_load_async_to_lds_b128(
          (AS1_ v4i*)gp, (AS3_ v4i*)&sB[buf][rB][cB], 0, 0);
#else
      *(float4*)&sB[buf][rB][cB] = *(const float4*)gp;
#endif
    } else {
      #pragma unroll
      for (int q = 0; q < 4; ++q)
        sB[buf][rB][cB + q] = (jj + q < NT_) ? gp[q] : 0.f;
    }
  };
  auto stage_fence = [&]() {
#if USE_ASYNC_LDS
    if (!tail) __builtin_amdgcn_s_wait_asynccnt(0);
#endif
    __syncthreads();
  };

  const int NCH = K2_ / KC_;           // 100 chunks
  stage(0, 0);
  stage_fence();
  for (int ch = 0; ch < NCH; ++ch) {
    int cur = ch & 1;
    if (ch + 1 < NCH) stage(cur ^ 1, (ch + 1) * KC_);
    const float* Ab = Gt + ((size_t)b * K2_ + (size_t)ch * KC_) * H3_ + wgM * 128 + mrow;
    #pragma unroll
    for (int ks = 0; ks < KC_; ks += 4) {
      v2f bb;
      bb.x = sB[cur][ks + koff][ngN * 16 + jcol];
      bb.y = sB[cur][ks + koff + 1][ngN * 16 + jcol];
      const float* Ar = Ab + (size_t)(ks + koff) * H3_;
      #pragma unroll
      for (int mt = 0; mt < 8; ++mt) {
        v2f a;
        a.x = Ar[mt * 16];
        a.y = Ar[mt * 16 + H3_];
        acc[mt] = wmma_f32_4(a, bb, acc[mt]);
      }
    }
    stage_fence();   // next buffer ready; all waves done reading cur
  }

  if (jg < NT_) {
    #pragma unroll
    for (int mt = 0; mt < 8; ++mt) {
      int rbase = wgM * 128 + mt * 16 + ((lane >> 4) << 3);
      #pragma unroll
      for (int v = 0; v < 8; ++v) {
        int o = rbase + v;
        float r = acc[mt][v] + b3d[o];
        fmap[((size_t)b * H3_ + o) * NT_ + jg] = r > 0.f ? r : 0.f;
      }
    }
  }
}

// ---------------------------------------------------------------
// 1x1 conv H3->H2 + ReLU as WMMA GEMM: M=128, N=10000, K=512 per batch.
// ---------------------------------------------------------------
__global__ __launch_bounds__(128) void k_gemm_c1(
    const float* __restrict__ w, const float* __restrict__ bias,
    const float* __restrict__ fmap, float* __restrict__ out) {
  const int MT = H2_ / 16;
  const int NTIL = NT_ / 16;
  int wave = threadIdx.x >> 5, lane = threadIdx.x & 31;
  int tile = blockIdx.x * 4 + wave;
  int nt = tile % NTIL; tile /= NTIL;
  int mt = tile % MT;   tile /= MT;
  int b  = tile;
  int o0 = mt * 16, j0 = nt * 16;
  int mrow = lane & 15;
  int koff = (lane >> 4) << 1;
  int jg = j0 + (lane & 15);
  v8f acc = {};
  const float* A  = w + (size_t)o0 * H3_;
  const float* Bm = fmap + (size_t)b * H3_ * NT_;
  for (int k0 = 0; k0 < H3_; k0 += 4) {
    v2f a, bb;
    const float* Ap = A + (size_t)mrow * H3_ + (k0 + koff);
    a.x = Ap[0];
    a.y = Ap[1];
    const float* Bp = Bm + (size_t)(k0 + koff) * NT_ + jg;
    bb.x = Bp[0];
    bb.y = Bp[NT_];
    acc = wmma_f32_4(a, bb, acc);
  }
  int rbase = o0 + ((lane >> 4) << 3);
  #pragma unroll
  for (int v = 0; v < 8; ++v) {
    int m = rbase + v;
    float r = acc[v] + bias[m];
    out[((size_t)b * H2_ + m) * NT_ + jg] = r > 0.f ? r : 0.f;
  }
}

// ---------------------------------------------------------------
// 3x3 conv H2->H2 (pad=1) + ReLU, LDS-tiled, weight-stationary.
// ---------------------------------------------------------------
#define CHK_ 32
__global__ __launch_bounds__(256) void k_conv3x3(
    const float* __restrict__ in, const float* __restrict__ w,
    const float* __restrict__ bias, float* __restrict__ out) {
  __shared__ float sIn[3][CHK_][T_ + 2];
  int bd = blockIdx.x;
  int d = bd % DM_, b = bd / DM_;
  int oc = threadIdx.x & 127;
  int sh = (threadIdx.x >> 7) * 50;
  float acc[50];
  #pragma unroll
  for (int i = 0; i < 50; ++i) acc[i] = bias[oc];
  for (int c0 = 0; c0 < H2_; c0 += CHK_) {
    __syncthreads();
    for (int idx = threadIdx.x; idx < 3 * CHK_ * (T_ + 2); idx += 256) {
      int s  = idx % (T_ + 2);
      int ic = (idx / (T_ + 2)) % CHK_;
      int dy = idx / ((T_ + 2) * CHK_);
      int dr = d + dy - 1;
      int ss = s - 1;
      float v = 0.f;
      if (dr >= 0 && dr < DM_ && ss >= 0 && ss < T_)
        v = in[((size_t)b * H2_ + (c0 + ic)) * NT_ + (size_t)dr * T_ + ss];
      sIn[dy][ic][s] = v;
    }
    __syncthreads();
    const float* wp = w + ((size_t)oc * H2_ + c0) * 9;
    for (int ic = 0; ic < CHK_; ++ic) {
      float w00 = wp[0], w01 = wp[1], w02 = wp[2];
      float w10 = wp[3], w11 = wp[4], w12 = wp[5];
      float w20 = wp[6], w21 = wp[7], w22 = wp[8];
      wp += 9;
      const float* r0 = &sIn[0][ic][sh];
      const float* r1 = &sIn[1][ic][sh];
      const float* r2 = &sIn[2][ic][sh];
      #pragma unroll
      for (int i = 0; i < 50; ++i) {
        acc[i] += w00 * r0[i] + w01 * r0[i + 1] + w02 * r0[i + 2]
                + w10 * r1[i] + w11 * r1[i + 1] + w12 * r1[i + 2]
                + w20 * r2[i] + w21 * r2[i + 1] + w22 * r2[i + 2];
      }
    }
  }
  #pragma unroll
  for (int i = 0; i < 50; ++i) {
    float v = acc[i];
    out[((size_t)b * H2_ + oc) * NT_ + (size_t)d * T_ + sh + i] = v > 0.f ? v : 0.f;
  }
}

// ---------------------------------------------------------------
// Final 1x1 conv H2->2 + sigmoid -> conf [B,2,DM,T].
// ---------------------------------------------------------------
__global__ __launch_bounds__(256) void k_c4(
    const float* __restrict__ in, const float* __restrict__ w,
    const float* __restrict__ bias, float* __restrict__ conf) {
  int idx = blockIdx.x * 256 + threadIdx.x;
  if (idx >= B_ * NT_) return;
  int j = idx % NT_, b = idx / NT_;
  float a0 = bias[0], a1 = bias[1];
  const float* ip = in + (size_t)b * H2_ * NT_ + j;
  for (int k = 0; k < H2_; ++k) {
    float v = ip[(size_t)k * NT_];
    a0 += w[k] * v;
    a1 += w[H2_ + k] * v;
  }
  conf[(size_t)b * 2 * NT_ + j]       = sigmoidf_(a0);
  conf[(size_t)b * 2 * NT_ + NT_ + j] = sigmoidf_(a1);
}

// ---------------------------------------------------------------
extern "C" void kernel_launch(void* const* d_in, const int* in_sizes, int n_in,
                              void* d_out, int out_size, void* d_ws, size_t ws_size,
                              hipStream_t stream) {
  const float* x    = (const float*)d_in[0];
  const float* w_b1 = (const float*)d_in[1];
  const float* b_b1 = (const float*)d_in[2];
  const float* w_b2 = (const float*)d_in[3];
  const float* b_b2 = (const float*)d_in[4];
  const float* w_s1 = (const float*)d_in[5];
  const float* b_s1 = (const float*)d_in[6];
  const float* w_s2 = (const float*)d_in[7];
  const float* b_s2 = (const float*)d_in[8];
  const float* w_e1 = (const float*)d_in[9];
  const float* b_e1 = (const float*)d_in[10];
  const float* w_e2 = (const float*)d_in[11];
  const float* b_e2 = (const float*)d_in[12];
  const float* w_p  = (const float*)d_in[13];
  const float* b_p  = (const float*)d_in[14];
  const float* w_3d = (const float*)d_in[15];
  const float* b_3d = (const float*)d_in[16];
  const float* w_c1 = (const float*)d_in[17];
  const float* b_c1 = (const float*)d_in[18];
  const float* w_c2 = (const float*)d_in[19];
  const float* b_c2 = (const float*)d_in[20];
  const float* w_c3 = (const float*)d_in[21];
  const float* b_c3 = (const float*)d_in[22];
  const float* w_c4 = (const float*)d_in[23];
  const float* b_c4 = (const float*)d_in[24];
  const float* sample_mask = (const float*)d_in[25];

  float* ws    = (float*)d_ws;
  float* h1    = ws;                 //    51,200
  float* h2    = ws + 51200;         //    51,200
  float* p_pad = ws + 102400;        //    57,344  (B,H1,TP_)
  float* Gt    = ws + 159744;        // 3,276,800  (B, K2_, H3_)  K-major
  float* c1o   = ws + 3436544;       // 2,560,000
  float* c2o   = ws + 5996544;       // 2,560,000
  float* c3o   = ws + 8556544;       // 2,560,000  (total ~44.5 MB)

  float* out    = (float*)d_out;     // conf | start | end | fmap
  float* conf   = out;
  float* startv = out + 40000;
  float* endv   = out + 40200;
  float* fmap   = out + 40400;

  // 1-D trunk + heads + proposal features (p zero-padded to TP_)
  k_zero<<<(B_ * H1_ * TP_ + 255) / 256, 256, 0, stream>>>(p_pad, B_ * H1_ * TP_);
  k_conv1d_grp_relu<<<B_ * T_, 256, 0, stream>>>(x,  w_b1, b_b1, h1, FEAT_ / 4);
  k_conv1d_grp_relu<<<B_ * T_, 256, 0, stream>>>(h1, w_b2, b_b2, h2, H1_ / 4);
  k_head<<<B_ * T_, 256, 0, stream>>>(h2, w_s1, b_s1, w_s2, b_s2, startv);
  k_head<<<B_ * T_, 256, 0, stream>>>(h2, w_e1, b_e1, w_e2, b_e2, endv);
  k_conv1d_dense_relu<<<B_ * T_, 256, 0, stream>>>(h2, w_p, b_p, p_pad);

  // Reassociated BM layer: two WMMA GEMMs (no 655 MB intermediate)
  k_gemm_g<<<(B_ * NS_ * (H3_ / 16) * 7) / 4, 128, 0, stream>>>(w_3d, p_pad, Gt);
  k_gemm_fmap<<<B_ * NJB_, 256, 0, stream>>>(Gt, sample_mask, b_3d, fmap);

  // 2-D confidence head
  k_gemm_c1<<<(B_ * (H2_ / 16) * (NT_ / 16)) / 4, 128, 0, stream>>>(w_c1, b_c1, fmap, c1o);
  k_conv3x3<<<B_ * DM_, 256, 0, stream>>>(c1o, w_c2, b_c2, c2o);
  k_conv3x3<<<B_ * DM_, 256, 0, stream>>>(c2o, w_c3, b_c3, c3o);
  k_c4<<<(B_ * NT_ + 255) / 256, 256, 0, stream>>>(c3o, w_c4, b_c4, conf);
}